// VectorGraphicsRGB2IntegrandSlang_34918084116543
// MI455X (gfx1250) — compile-verified
//
#include <hip/hip_runtime.h>

#define NPTS 2097152
#define NTILES 13   // 26 padded cells / 2 cells-per-tile (cell 25 is zero padding)

typedef __attribute__((ext_vector_type(2))) float v2f;
typedef __attribute__((ext_vector_type(8))) float v8f;

// Coefficient of feature k for padded-test (cell, v).
// Features: k0=px, k1=py, k2=px^2, k3=py^2, k4=1, k5..7=0.
// Tests per cell: v0..2 = triangle edges, v3 = circle q, v4 = ellipse q.
__device__ __forceinline__ float coef_value(const float* __restrict__ p,
                                            int cell, int v, int k) {
    if (cell >= 25 || v > 4 || k > 4) return 0.0f;
    const float* cp = p + cell * 28;
    if (v < 3) {  // triangle edge: e = px*(y1-y0) - py*(x1-x0) + (y0*(x1-x0) - x0*(y1-y0))
        float x0, y0, x1, y1;
        if (v == 0)      { x0 = cp[1]; y0 = cp[2]; x1 = cp[3]; y1 = cp[4]; }
        else if (v == 1) { x0 = cp[3]; y0 = cp[4]; x1 = cp[5]; y1 = cp[6]; }
        else             { x0 = cp[5]; y0 = cp[6]; x1 = cp[1]; y1 = cp[2]; }
        if (k == 0) return y1 - y0;
        if (k == 1) return -(x1 - x0);
        if (k == 4) return y0 * (x1 - x0) - x0 * (y1 - y0);
        return 0.0f;
    } else if (v == 3) {  // circle: inside iff q <= 0
        float cx = cp[12], cy = cp[13], r = cp[14];
        if (k == 0) return -2.0f * cx;
        if (k == 1) return -2.0f * cy;
        if (k == 2) return 1.0f;
        if (k == 3) return 1.0f;
        if (k == 4) return cx * cx + cy * cy - r * r;
        return 0.0f;
    } else {              // ellipse: inside iff q <= 0
        float cx = cp[20], cy = cp[21], rx = cp[22], ry = cp[23];
        float ia = 1.0f / (rx * rx), ib = 1.0f / (ry * ry);
        if (k == 0) return -2.0f * cx * ia;
        if (k == 1) return -2.0f * cy * ib;
        if (k == 2) return ia;
        if (k == 3) return ib;
        if (k == 4) return cx * cx * ia + cy * cy * ib - 1.0f;
        return 0.0f;
    }
}

__global__ __launch_bounds__(256)
void vg_rgb_wmma_kernel(const float* __restrict__ x,
                        const float* __restrict__ p,
                        float* __restrict__ out) {
    // A-operand table in exact WMMA A-matrix (16x4 f32) register layout:
    // entry e = (tile, kchunk, lane); lane L supplies row M=L%16, K = 2*(L/16)+{0,1}.
    __shared__ float ldsA[NTILES * 2 * 32 * 2];
    __shared__ float ldsC[75 * 3];

    const int tid = threadIdx.x;
    for (int e = tid; e < NTILES * 2 * 32; e += 256) {
        int L     = e & 31;
        int kc    = (e >> 5) & 1;
        int t     = e >> 6;
        int Mrow  = L & 15;
        int hA    = L >> 4;
        int ptIdx = 16 * t + Mrow;      // padded-test row within the GEMM
        int cell  = ptIdx >> 3;
        int v     = ptIdx & 7;
        int kb    = kc * 4 + 2 * hA;
        ldsA[e * 2 + 0] = coef_value(p, cell, v, kb);
        ldsA[e * 2 + 1] = coef_value(p, cell, v, kb + 1);
    }
    if (tid < 75) {
        int cell = tid / 3, kind = tid % 3;
        const float* cp  = p + cell * 28;
        const float* col = (kind == 0) ? (cp + 8) : (kind == 1) ? (cp + 16) : (cp + 25);
        ldsC[tid * 3 + 0] = col[0];
        ldsC[tid * 3 + 1] = col[1];
        ldsC[tid * 3 + 2] = col[2];
    }
    __syncthreads();

    const int lane = tid & 31;
    const int wave = tid >> 5;
    const int h    = lane >> 4;   // half-wave: selects even (0) / odd (1) cell of a tile
    const int n    = lane & 15;   // point slot within the 16-point wave tile

    // Preload loop-invariant A operands into registers (26 x v2f).
    v2f A0[NTILES], A1[NTILES];
#pragma unroll
    for (int t = 0; t < NTILES; ++t) {
        const float* a0 = &ldsA[((t * 2 + 0) * 32 + lane) * 2];
        const float* a1 = &ldsA[((t * 2 + 1) * 32 + lane) * 2];
        A0[t].x = a0[0]; A0[t].y = a0[1];
        A1[t].x = a1[0]; A1[t].y = a1[1];
    }

    // Constant B for K-chunk 1: rows [1, 0] (lanes 0-15 = K4, K5) / [0, 0] (K6, K7).
    v2f Bc;
    Bc.x = (h == 0) ? 1.0f : 0.0f;
    Bc.y = 0.0f;

    const float2* x2 = (const float2*)x;

    for (int base = blockIdx.x * 128; base < NPTS; base += gridDim.x * 128) {
        const int P = base + wave * 16 + n;
        float2 pt = x2[P];
        float px = pt.x, py = pt.y;

        // B for K-chunk 0: lanes 0-15 hold rows K0(px),K1(py); lanes 16-31 rows K2(px^2),K3(py^2)
        v2f B0;
        B0.x = h ? px * px : px;
        B0.y = h ? py * py : py;

        int best = -1;
#pragma unroll
        for (int t = 0; t < NTILES; ++t) {
            v8f acc = {};
            acc = __builtin_amdgcn_wmma_f32_16x16x4_f32(false, A0[t], false, B0,
                                                        (short)0, acc, false, false);
            acc = __builtin_amdgcn_wmma_f32_16x16x4_f32(false, A1[t], false, Bc,
                                                        (short)0, acc, false, false);
            // This lane holds cell 2t+h completely: acc[v] = test v of that cell.
            int cell = 2 * t + h;
            if (cell < 25) {
                float e0 = acc[0], e1 = acc[1], e2 = acc[2];
                float qc = acc[3], qe = acc[4];
                bool tri = ((e0 >= 0.f) & (e1 >= 0.f) & (e2 >= 0.f)) |
                           ((e0 <= 0.f) & (e1 <= 0.f) & (e2 <= 0.f));
                if (tri)        best = 3 * cell;
                if (qc <= 0.f)  best = 3 * cell + 1;
                if (qe <= 0.f)  best = 3 * cell + 2;
            }
        }
        // Merge even-cell half with odd-cell half of the wave.
        int other = __shfl_xor(best, 16);
        best = max(best, other);

        if (h == 0) {
            float r = 0.f, g = 0.f, b = 0.f;
            if (best >= 0) {
                r = ldsC[best * 3 + 0];
                g = ldsC[best * 3 + 1];
                b = ldsC[best * 3 + 2];
            }
            size_t o = (size_t)P * 3;
            out[o + 0] = r;
            out[o + 1] = g;
            out[o + 2] = b;
        }
    }
}

extern "C" void kernel_launch(void* const* d_in, const int* in_sizes, int n_in,
                              void* d_out, int out_size, void* d_ws, size_t ws_size,
                              hipStream_t stream) {
    (void)in_sizes; (void)n_in; (void)d_ws; (void)ws_size; (void)out_size;
    const float* x = (const float*)d_in[0];   // (2097152, 2) f32
    const float* p = (const float*)d_in[1];   // (700,) f32
    float* out = (float*)d_out;               // (2097152, 3) f32
    // 2048 blocks * 128 points/block/iter * 8 iters = 2,097,152 points exactly.
    vg_rgb_wmma_kernel<<<2048, 256, 0, stream>>>(x, p, out);
}